// ParallelQuantumLayer_41446434406847
// MI455X (gfx1250) — compile-verified
//
#include <hip/hip_runtime.h>
#include <hip/hip_bf16.h>
#include <math.h>

#define N_QUBITS   10
#define N_CIRCUITS 16
#define DEPTH      2
#define IN_F       1568
#define BATCH      256
#define TOTAL_F    (N_CIRCUITS * N_QUBITS)   // 160
#define BN_EPS     1e-5f
#define PI_F       3.14159265358979323846f

typedef __attribute__((ext_vector_type(2))) float v2f;
typedef __attribute__((ext_vector_type(8))) float v8f;

// ---------------------------------------------------------------------------
// Kernel 1: fused Linear  h = x @ W + b  using full-precision f32 WMMA.
// One wave per 16x16 output tile; K loop of 392 x V_WMMA_F32_16X16X4_F32.
// A (16x4 f32) layout: lanes 0-15 hold M=lane; v0 = K=0|K=2, v1 = K=1|K=3.
// B (4x16 f32) layout mirrors: lanes give N; v0 = K=0|K=2, v1 = K=1|K=3.
// C/D: VGPR r, lanes 0-15 -> M=r, lanes 16-31 -> M=8+r, N=lane&15.
// ---------------------------------------------------------------------------
__global__ __launch_bounds__(32) void gemm_wmma_f32(
    const float* __restrict__ x,   // (256, 1568)
    const float* __restrict__ W,   // (1568, 160)
    const float* __restrict__ b,   // (160,)
    float* __restrict__ h)         // (256, 160)
{
    const int row0 = blockIdx.x * 16;       // 16 M-tiles
    const int col0 = blockIdx.y * 16;       // 10 N-tiles
    const int lane = threadIdx.x & 31;
    const int half = lane >> 4;             // 0 or 1
    const int m    = lane & 15;

    v8f acc = {};
    const float* arow = x + (row0 + m) * IN_F;
    for (int k = 0; k < IN_F; k += 4) {
        v2f a, bm;
        a.x  = arow[k + 2 * half + 0];
        a.y  = arow[k + 2 * half + 1];
        bm.x = W[(k + 2 * half + 0) * TOTAL_F + col0 + m];
        bm.y = W[(k + 2 * half + 1) * TOTAL_F + col0 + m];
        acc = __builtin_amdgcn_wmma_f32_16x16x4_f32(
            /*neg_a=*/false, a, /*neg_b=*/false, bm,
            /*c_mod=*/(short)0, acc, /*reuse_a=*/false, /*reuse_b=*/false);
    }
    const float bias = b[col0 + m];
#pragma unroll
    for (int r = 0; r < 8; ++r) {
        const int row = row0 + r + 8 * half;
        h[row * TOTAL_F + col0 + m] = acc[r] + bias;
    }
}

// ---------------------------------------------------------------------------
// Kernel 2: BatchNorm1d (training batch stats, biased var) + ReLU + *pi.
// One block per feature column; 256 threads (one per batch row).
// ---------------------------------------------------------------------------
__global__ __launch_bounds__(256) void bn_relu_pi(
    const float* __restrict__ h,      // (256, 160)
    const float* __restrict__ gamma,  // (160,)
    const float* __restrict__ beta,   // (160,)
    float* __restrict__ ang)          // (256, 160)  angles = relu(bn(h)) * pi
{
    const int col = blockIdx.x;
    const int t   = threadIdx.x;
    const float v = h[t * TOTAL_F + col];

    float s1 = v, s2 = v * v;
#pragma unroll
    for (int off = 16; off > 0; off >>= 1) {
        s1 += __shfl_xor(s1, off, 32);
        s2 += __shfl_xor(s2, off, 32);
    }
    __shared__ float ls1[8], ls2[8];
    const int w = t >> 5, lane = t & 31;
    if (lane == 0) { ls1[w] = s1; ls2[w] = s2; }
    __syncthreads();
    float S1 = 0.f, S2 = 0.f;
#pragma unroll
    for (int i = 0; i < 8; ++i) { S1 += ls1[i]; S2 += ls2[i]; }

    const float mean = S1 * (1.0f / BATCH);
    const float var  = S2 * (1.0f / BATCH) - mean * mean;
    const float inv  = rsqrtf(var + BN_EPS);
    float o = (v - mean) * inv * gamma[col] + beta[col];
    o = fmaxf(o, 0.0f) * PI_F;
    ang[t * TOTAL_F + col] = o;
}

// ---------------------------------------------------------------------------
// Kernel 3: 10-qubit statevector sim, one wave32 per (circuit, sample).
// Amplitude flat index i (10 bits): lane = i>>5 (bits 9..5), reg = i&31.
// Wire w lives at flat bit (9-w):  w<=4 -> lane bit (4-w),  w>=5 -> reg bit (9-w).
// ---------------------------------------------------------------------------
__device__ __forceinline__ float2 cmul(float2 a, float2 b) {
    return make_float2(a.x * b.x - a.y * b.y, a.x * b.y + a.y * b.x);
}
__device__ __forceinline__ float2 cadd(float2 a, float2 b) {
    return make_float2(a.x + b.x, a.y + b.y);
}
__device__ __forceinline__ float2 shfl_xor2(float2 v, int mask) {
    float2 r;
    r.x = __shfl_xor(v.x, mask, 32);
    r.y = __shfl_xor(v.y, mask, 32);
    return r;
}

// General 1-qubit gate [[u00,u01],[u10,u11]] on wire WIRE.
template <int WIRE>
__device__ __forceinline__ void apply_u3(float2 s[32], float2 u00, float2 u01,
                                         float2 u10, float2 u11, int lane) {
    if constexpr (WIRE >= 5) {                      // register-bit wire
        constexpr int m = 1 << (9 - WIRE);
#pragma unroll
        for (int r = 0; r < 32; ++r) {
            if ((r & m) == 0) {
                const float2 a0 = s[r], a1 = s[r | m];
                s[r]     = cadd(cmul(u00, a0), cmul(u01, a1));
                s[r | m] = cadd(cmul(u10, a0), cmul(u11, a1));
            }
        }
    } else {                                        // lane-bit wire
        constexpr int mask = 1 << (4 - WIRE);
        const bool hi = (lane & mask) != 0;
        const float2 cm = hi ? u11 : u00;           // coeff on my amplitude
        const float2 co = hi ? u10 : u01;           // coeff on partner amplitude
#pragma unroll
        for (int r = 0; r < 32; ++r) {
            const float2 o = shfl_xor2(s[r], mask);
            s[r] = cadd(cmul(cm, s[r]), cmul(co, o));
        }
    }
}

// RX embedding: [[c, -i s], [-i s, c]], half angle.
template <int WIRE>
__device__ __forceinline__ void apply_rx(float2 s[32], float theta, int lane) {
    const float hh = theta * 0.5f;
    const float c = __cosf(hh) * 0.f + cosf(hh);    // keep exact libm cosf
    const float sn = sinf(hh);
    const float2 u00 = make_float2(c, 0.f);
    const float2 u01 = make_float2(0.f, -sn);
    apply_u3<WIRE>(s, u00, u01, u01, u00, lane);
}

// Fused RZ(t2)*RY(t1)*RZ(t0) (applied in reference order RZ,RY,RZ).
template <int WIRE>
__device__ __forceinline__ void apply_rot(float2 s[32], float t0, float t1,
                                          float t2, int lane) {
    const float P = 0.5f * (t2 + t0);
    const float M = 0.5f * (t2 - t0);
    const float c = cosf(0.5f * t1), sn = sinf(0.5f * t1);
    const float cp = cosf(P), sp = sinf(P);
    const float cmm = cosf(M), smm = sinf(M);
    const float2 u00 = make_float2(c * cp, -c * sp);   //  c * e^{-iP}
    const float2 u01 = make_float2(-sn * cmm, sn * smm); // -s * e^{-iM}
    const float2 u10 = make_float2(sn * cmm, sn * smm);  //  s * e^{+iM}
    const float2 u11 = make_float2(c * cp, c * sp);      //  c * e^{+iP}
    apply_u3<WIRE>(s, u00, u01, u10, u11, lane);
}

template <int CW, int TW>
__device__ __forceinline__ void cnot(float2 s[32], int lane) {
    if constexpr (CW >= 5 && TW >= 5) {             // both register bits: rename
        constexpr int cm = 1 << (9 - CW), tm = 1 << (9 - TW);
#pragma unroll
        for (int r = 0; r < 32; ++r) {
            if ((r & cm) && !(r & tm)) {
                const float2 t = s[r]; s[r] = s[r | tm]; s[r | tm] = t;
            }
        }
    } else if constexpr (CW >= 5 && TW <= 4) {      // ctrl reg, tgt lane: shuffle
        constexpr int cm = 1 << (9 - CW);
        constexpr int lm = 1 << (4 - TW);
#pragma unroll
        for (int r = 0; r < 32; ++r)
            if (r & cm) s[r] = shfl_xor2(s[r], lm);
    } else if constexpr (CW <= 4 && TW >= 5) {      // ctrl lane, tgt reg: cndmask
        constexpr int clm = 1 << (4 - CW);
        constexpr int tm  = 1 << (9 - TW);
        const bool act = (lane & clm) != 0;
#pragma unroll
        for (int r = 0; r < 32; ++r) {
            if ((r & tm) == 0) {
                const float2 a = s[r], bb = s[r | tm];
                s[r]      = act ? bb : a;
                s[r | tm] = act ? a  : bb;
            }
        }
    } else {                                        // both lane bits
        constexpr int clm = 1 << (4 - CW);
        constexpr int tlm = 1 << (4 - TW);
        const bool act = (lane & clm) != 0;
#pragma unroll
        for (int r = 0; r < 32; ++r) {
            const float2 o = shfl_xor2(s[r], tlm);
            s[r] = act ? o : s[r];
        }
    }
}

// Per-lane partial such that <Y_w> = sum over all 32 lanes of the partial.
template <int WIRE>
__device__ __forceinline__ float expval_partial(const float2 s[32], int lane) {
    float p = 0.f;
    if constexpr (WIRE >= 5) {
        constexpr int m = 1 << (9 - WIRE);
#pragma unroll
        for (int r = 0; r < 32; ++r) {
            if ((r & m) == 0) {
                const float2 a = s[r], bb = s[r | m];
                p += a.x * bb.y - a.y * bb.x;        // Im(conj(a0)*a1)
            }
        }
        p *= 2.0f;
    } else {
        constexpr int mask = 1 << (4 - WIRE);
        const bool hi = (lane & mask) != 0;
#pragma unroll
        for (int r = 0; r < 32; ++r) {
            const float2 o  = shfl_xor2(s[r], mask);
            const float2 a0 = hi ? o : s[r];
            const float2 a1 = hi ? s[r] : o;
            p += a0.x * a1.y - a0.y * a1.x;          // each pair counted twice == *2
        }
    }
    return p;
}

__global__ __launch_bounds__(256) void qsim(
    const float* __restrict__ ang,   // (256, 160) angles (already * pi)
    const float* __restrict__ wts,   // (16, 60) = (circ, layer*30 + rot*10 + q)
    float* __restrict__ out)         // (256, 160)
{
    const int wave = threadIdx.x >> 5;
    const int lane = threadIdx.x & 31;
    const int sim  = blockIdx.x * 8 + wave;   // 0..4095
    const int c    = sim >> 8;                // circuit
    const int bidx = sim & 255;               // batch sample

    const float* th = ang + bidx * TOTAL_F + c * N_QUBITS;
    const float* wc = wts + c * (DEPTH * 3 * N_QUBITS);

    float2 s[32];
#pragma unroll
    for (int r = 0; r < 32; ++r) s[r] = make_float2(0.f, 0.f);
    if (lane == 0) s[0] = make_float2(1.f, 0.f);   // |0...0>

    // --- RX embedding on all 10 wires ---
    apply_rx<0>(s, th[0], lane); apply_rx<1>(s, th[1], lane);
    apply_rx<2>(s, th[2], lane); apply_rx<3>(s, th[3], lane);
    apply_rx<4>(s, th[4], lane); apply_rx<5>(s, th[5], lane);
    apply_rx<6>(s, th[6], lane); apply_rx<7>(s, th[7], lane);
    apply_rx<8>(s, th[8], lane); apply_rx<9>(s, th[9], lane);

    // --- DEPTH layers of fused RZ*RY*RZ followed by CNOT ring ---
    for (int layer = 0; layer < DEPTH; ++layer) {
        const float* wl = wc + layer * 3 * N_QUBITS;
        apply_rot<0>(s, wl[0], wl[10], wl[20], lane);
        apply_rot<1>(s, wl[1], wl[11], wl[21], lane);
        apply_rot<2>(s, wl[2], wl[12], wl[22], lane);
        apply_rot<3>(s, wl[3], wl[13], wl[23], lane);
        apply_rot<4>(s, wl[4], wl[14], wl[24], lane);
        apply_rot<5>(s, wl[5], wl[15], wl[25], lane);
        apply_rot<6>(s, wl[6], wl[16], wl[26], lane);
        apply_rot<7>(s, wl[7], wl[17], wl[27], lane);
        apply_rot<8>(s, wl[8], wl[18], wl[28], lane);
        apply_rot<9>(s, wl[9], wl[19], wl[29], lane);

        cnot<0, 1>(s, lane); cnot<1, 2>(s, lane); cnot<2, 3>(s, lane);
        cnot<3, 4>(s, lane); cnot<4, 5>(s, lane); cnot<5, 6>(s, lane);
        cnot<6, 7>(s, lane); cnot<7, 8>(s, lane); cnot<8, 9>(s, lane);
        cnot<9, 0>(s, lane);
    }

    // --- <Y_w> for all 10 wires ---
    float ev[10];
    ev[0] = expval_partial<0>(s, lane); ev[1] = expval_partial<1>(s, lane);
    ev[2] = expval_partial<2>(s, lane); ev[3] = expval_partial<3>(s, lane);
    ev[4] = expval_partial<4>(s, lane); ev[5] = expval_partial<5>(s, lane);
    ev[6] = expval_partial<6>(s, lane); ev[7] = expval_partial<7>(s, lane);
    ev[8] = expval_partial<8>(s, lane); ev[9] = expval_partial<9>(s, lane);

#pragma unroll
    for (int q = 0; q < 10; ++q) {
#pragma unroll
        for (int off = 16; off > 0; off >>= 1)
            ev[q] += __shfl_xor(ev[q], off, 32);
    }
    if (lane == 0) {
        float* o = out + bidx * TOTAL_F + c * N_QUBITS;
#pragma unroll
        for (int q = 0; q < 10; ++q) o[q] = ev[q];
    }
}

// ---------------------------------------------------------------------------
extern "C" void kernel_launch(void* const* d_in, const int* in_sizes, int n_in,
                              void* d_out, int out_size, void* d_ws, size_t ws_size,
                              hipStream_t stream) {
    (void)in_sizes; (void)n_in; (void)out_size; (void)ws_size;
    const float* x     = (const float*)d_in[0];   // (256, 1568)
    const float* W     = (const float*)d_in[1];   // (1568, 160)
    const float* b     = (const float*)d_in[2];   // (160,)
    const float* gamma = (const float*)d_in[3];   // (160,)
    const float* beta  = (const float*)d_in[4];   // (160,)
    const float* wts   = (const float*)d_in[5];   // (16, 60)
    float* out = (float*)d_out;                   // (256, 160)

    float* h   = (float*)d_ws;                    // (256, 160)
    float* ang = h + BATCH * TOTAL_F;             // (256, 160)

    dim3 g1(BATCH / 16, TOTAL_F / 16);            // 16 x 10 tiles
    gemm_wmma_f32<<<g1, 32, 0, stream>>>(x, W, b, h);
    bn_relu_pi<<<TOTAL_F, BATCH, 0, stream>>>(h, gamma, beta, ang);
    qsim<<<(N_CIRCUITS * BATCH) / 8, 256, 0, stream>>>(ang, wts, out);
}